// LowRankRayTracer_27599459844410
// MI455X (gfx1250) — compile-verified
//
#include <hip/hip_runtime.h>

// LowRankRayTracer CSI kernel for gfx1250 (MI455X).
// Strategy: fused complex GEMM (V_WMMA_F32_16X16X4_F32) + in-register cumsum
// epilogue + deterministic two-stage reduction through d_ws.

typedef __attribute__((ext_vector_type(2))) float v2f;
typedef __attribute__((ext_vector_type(8))) float v8f;

#define DIRS  512
#define PTS   64
#define RANK  32
#define NFREQ 256

__device__ __forceinline__ v8f wmma_f32(v2f a, v2f b, v8f c) {
  // D = A(16x4 f32) * B(4x16 f32) + C(16x16 f32)
  return __builtin_amdgcn_wmma_f32_16x16x4_f32(
      /*neg_a=*/false, a, /*neg_b=*/false, b,
      /*c_mod=*/(short)0, c, /*reuse_a=*/false, /*reuse_b=*/false);
}

__global__ __launch_bounds__(256) void csi_main_kernel(
    const float* __restrict__ att_re, const float* __restrict__ att_im,
    const float* __restrict__ rad_re, const float* __restrict__ rad_im,
    const float* __restrict__ fre,    const float* __restrict__ fim,
    const int* __restrict__ mrl, float2* __restrict__ ws)
{
  const int d    = blockIdx.x;           // direction
  const int lane = threadIdx.x & 31;
  const int wv   = threadIdx.x >> 5;     // 8 waves per block
  const int l15  = lane & 15;
  const int lh   = lane >> 4;            // 0: rows 0-7 / K 0-1,  1: rows 8-15 / K 2-3
  const float dt = (float)(*mrl) / (float)PTS;

  // cumsum carry + column sums, one per owned frequency column-tile
  float carry_re[2] = {0.f, 0.f}, carry_im[2] = {0.f, 0.f};
  float col_re [2] = {0.f, 0.f}, col_im [2] = {0.f, 0.f};

  for (int m = 0; m < 4; ++m) {          // row tiles over P=64
    // A fragments: lane -> row = m*16 + l15, k = 4*kk + 2*lh + {0,1}
    const long rowbase = ((long)d * PTS + m * 16 + l15) * RANK + 2 * lh;
    v2f ar[8], ai[8], rr[8], ri[8];
    #pragma unroll
    for (int kk = 0; kk < 8; ++kk) {
      ar[kk] = *(const v2f*)(att_re + rowbase + 4 * kk);
      ai[kk] = *(const v2f*)(att_im + rowbase + 4 * kk);
      rr[kk] = *(const v2f*)(rad_re + rowbase + 4 * kk);
      ri[kk] = *(const v2f*)(rad_im + rowbase + 4 * kk);
    }

    #pragma unroll
    for (int ni = 0; ni < 2; ++ni) {
      const int n = wv * 2 + ni;         // frequency column-tile
      // B fragments: lane -> col f = n*16 + l15, k = 4*kk + 2*lh + {0,1}
      const long fbase = (long)(n * 16 + l15) * RANK + 2 * lh;

      v8f X1 = (v8f)0.f, X2 = (v8f)0.f, X3 = (v8f)0.f, X4 = (v8f)0.f;
      v8f Y1 = (v8f)0.f, Y2 = (v8f)0.f, Y3 = (v8f)0.f, Y4 = (v8f)0.f;
      #pragma unroll
      for (int kk = 0; kk < 8; ++kk) {   // K = 32 in steps of 4
        v2f br = *(const v2f*)(fre + fbase + 4 * kk);
        v2f bi = *(const v2f*)(fim + fbase + 4 * kk);
        X1 = wmma_f32(ar[kk], br, X1);   // att_re . f_re
        X2 = wmma_f32(ai[kk], bi, X2);   // att_im . f_im
        X3 = wmma_f32(ar[kk], bi, X3);   // att_re . f_im
        X4 = wmma_f32(ai[kk], br, X4);   // att_im . f_re
        Y1 = wmma_f32(rr[kk], br, Y1);   // rad_re . f_re
        Y2 = wmma_f32(ri[kk], bi, Y2);   // rad_im . f_im
        Y3 = wmma_f32(rr[kk], bi, Y3);   // rad_re . f_im
        Y4 = wmma_f32(ri[kk], br, Y4);   // rad_im . f_re
      }

      // a_proj = (X1+X2) + i(X3-X4)   (conj att . fv)
      // t_proj = (X1-X2) + i(X3+X4)   (att . fv)  -> cumsum -> u_proj
      // r_proj = (Y1+Y2) + i(Y3-Y4)   (conj rad . fv)
      float tre[8], tim[8], gre[8], gim[8], hre[8], him[8];
      #pragma unroll
      for (int v = 0; v < 8; ++v) {
        tre[v] = X1[v] - X2[v]; tim[v] = X3[v] + X4[v];
        gre[v] = X1[v] + X2[v]; gim[v] = X3[v] - X4[v];
        hre[v] = Y1[v] + Y2[v]; him[v] = Y3[v] - Y4[v];
      }

      // half-tile total of t (rows 0-7 in lanes 0-15, rows 8-15 in lanes 16-31)
      float s_re = 0.f, s_im = 0.f;
      #pragma unroll
      for (int v = 0; v < 8; ++v) { s_re += tre[v]; s_im += tim[v]; }

      // high half needs the low half's total for its prefix base
      float low_re = __shfl(s_re, l15, 32);
      float low_im = __shfl(s_im, l15, 32);
      float run_re = carry_re[ni] + (lh ? low_re : 0.f);
      float run_im = carry_im[ni] + (lh ? low_im : 0.f);

      float cs_re = col_re[ni], cs_im = col_im[ni];
      #pragma unroll
      for (int v = 0; v < 8; ++v) {
        run_re += tre[v]; run_im += tim[v];       // inclusive cumsum
        float u_re = dt * run_re, u_im = dt * run_im;
        float ra_re = hre[v] * gre[v] - him[v] * gim[v];
        float ra_im = hre[v] * gim[v] + him[v] * gre[v];
        int p = m * 16 + lh * 8 + v;
        float w_re = (p == 0) ? 1.f : (1.f + u_re);  // p=0: first-order only
        float w_im = (p == 0) ? 0.f : u_im;
        cs_re += ra_re * w_re - ra_im * w_im;
        cs_im += ra_re * w_im + ra_im * w_re;
      }
      col_re[ni] = cs_re; col_im[ni] = cs_im;

      // carry for next row tile: full 16-row total of t
      carry_re[ni] += s_re + __shfl_xor(s_re, 16, 32);
      carry_im[ni] += s_im + __shfl_xor(s_im, 16, 32);
    }
  }

  // combine the two row halves per column and emit per-direction partials
  const float scale = dt / (float)DIRS;
  #pragma unroll
  for (int ni = 0; ni < 2; ++ni) {
    float tr = col_re[ni] + __shfl_xor(col_re[ni], 16, 32);
    float ti = col_im[ni] + __shfl_xor(col_im[ni], 16, 32);
    if (lh == 0) {
      int f = (wv * 2 + ni) * 16 + l15;
      ws[(size_t)f * DIRS + d] = make_float2(tr * scale, ti * scale);
    }
  }
}

// Deterministic final reduction over directions (fixed order, no atomics).
__global__ __launch_bounds__(256) void csi_reduce_kernel(
    const float2* __restrict__ ws, float* __restrict__ out)
{
  int f = threadIdx.x;                   // 256 frequencies
  float sr = 0.f, si = 0.f;
  for (int d = 0; d < DIRS; ++d) {
    float2 v = ws[(size_t)f * DIRS + d];
    sr += v.x; si += v.y;
  }
  out[2 * f]     = sr;                   // complex64 interleaved
  out[2 * f + 1] = si;
}

extern "C" void kernel_launch(void* const* d_in, const int* in_sizes, int n_in,
                              void* d_out, int out_size, void* d_ws, size_t ws_size,
                              hipStream_t stream) {
  const float* att_re = (const float*)d_in[0];
  const float* att_im = (const float*)d_in[1];
  const float* rad_re = (const float*)d_in[2];
  const float* rad_im = (const float*)d_in[3];
  const float* fre    = (const float*)d_in[4];
  const float* fim    = (const float*)d_in[5];
  const int*   mrl    = (const int*)d_in[6];

  float2* ws  = (float2*)d_ws;           // needs 256*512*8 = 1 MB
  float*  out = (float*)d_out;

  csi_main_kernel<<<DIRS, 256, 0, stream>>>(att_re, att_im, rad_re, rad_im,
                                            fre, fim, mrl, ws);
  csi_reduce_kernel<<<1, 256, 0, stream>>>(ws, out);
}